// QuantizedEmbedding_6717328851395
// MI455X (gfx1250) — compile-verified
//
#include <hip/hip_runtime.h>

// QuantizedEmbedding dequant-gather for MI455X (gfx1250, wave32).
//
// out[t, :] = float(weight_q[x[t], :]) * scales[x[t]]
//   tokens  = BATCH*SEQ = 16384, DIM = 512, VOCAB = 128000
//
// Memory-bound: ~67 MB moved -> ~2.9 us at 23.3 TB/s. Strategy:
//   - 128 threads (4 wave32) per token-row: 1x global_load_b128 +
//     1x global_store_b128 (nontemporal) per lane.
//   - Grid-stride software pipeline: load next token's index early and
//     global_prefetch_b8 its weight row to hide the dependent
//     index->row latency chain.
//   - Block-uniform index/scale -> scalar (SMEM) loads.

typedef int   __attribute__((ext_vector_type(4))) int4v;
typedef float __attribute__((ext_vector_type(4))) float4v;

#define QE_DIM      512
#define QE_THREADS  128   // QE_DIM / 4 elems-per-lane
#define QE_BLOCKS   2048  // 16384 tokens -> 8 pipelined iterations per block

__global__ __launch_bounds__(QE_THREADS) void qembed_dequant_gather(
    const int*   __restrict__ x,       // [n_tokens] row indices (int32)
    const int*   __restrict__ wq,      // [VOCAB, DIM] int8-valued weights in int32
    const float* __restrict__ scales,  // [VOCAB]
    float*       __restrict__ out,     // [n_tokens, DIM]
    int n_tokens)
{
    const int lane   = threadIdx.x;          // 0..127
    const int stride = gridDim.x;

    int t = blockIdx.x;
    if (t >= n_tokens) return;

    // Prologue of the software pipeline: index of the first token.
    int idx = x[t];

    while (t < n_tokens) {
        const int next_t = t + stride;
        int next_idx = 0;

        if (next_t < n_tokens) {
            next_idx = x[next_t];
            // Prefetch next row into cache while we chew on the current one.
            // Each lane prefetches the start of its 16B granule; lowers to
            // global_prefetch_b8 (gfx1250). Speculative-OK: address is valid.
            const char* p = (const char*)(wq + (size_t)next_idx * QE_DIM)
                            + (size_t)lane * 16;
            __builtin_prefetch(p, /*rw=*/0, /*locality=*/1);
        }

        // Block-uniform scalar loads.
        const float s = scales[idx];

        // One 128-bit load per lane covers the whole 2KB row per block.
        const int4v* row = (const int4v*)(wq + (size_t)idx * QE_DIM);
        int4v w = row[lane];                               // global_load_b128

        float4v f = __builtin_convertvector(w, float4v) * s; // v_cvt + v_mul

        // Streaming output: nontemporal store keeps the 33MB result from
        // evicting reusable weight rows out of the 192MB L2.
        float4v* orow = (float4v*)(out + (size_t)t * QE_DIM);
        __builtin_nontemporal_store(f, orow + lane);       // global_store_b128 nt

        t   = next_t;
        idx = next_idx;
    }
}

extern "C" void kernel_launch(void* const* d_in, const int* in_sizes, int n_in,
                              void* d_out, int out_size, void* d_ws, size_t ws_size,
                              hipStream_t stream) {
    const int*   x      = (const int*)  d_in[0];  // [BATCH*SEQ] indices
    const int*   wq     = (const int*)  d_in[1];  // [VOCAB*DIM]
    const float* scales = (const float*)d_in[2];  // [VOCAB]
    float*       out    = (float*)      d_out;    // [BATCH*SEQ*DIM] f32

    const int n_tokens = in_sizes[0];             // 16384

    int blocks = QE_BLOCKS;
    if (blocks > n_tokens) blocks = n_tokens;     // never launch idle blocks

    qembed_dequant_gather<<<blocks, QE_THREADS, 0, stream>>>(
        x, wq, scales, out, n_tokens);
}